// gradient_clipping_82463372083920
// MI455X (gfx1250) — compile-verified
//
#include <hip/hip_runtime.h>
#include <hip/hip_bf16.h>
#include <math.h>

// ---------------------------------------------------------------------------
// Global L2-norm gradient clipping for MI455X (gfx1250).
// Bandwidth-bound: ~2.43 GB of HBM traffic => ~104us floor at 23.3 TB/s.
// Pass 1: sum of squares via async global->LDS (ASYNCcnt) double-buffered
//         per-wave staging, wave32 shfl reduction, f64 global atomic combine.
// Pass 2: NT-hinted b128 load / scale / NT b128 store, with global prefetch.
// ---------------------------------------------------------------------------

typedef float v4f __attribute__((ext_vector_type(4)));

#define RED_BLOCK 256
#define RED_WAVES (RED_BLOCK / 32)

// -------------------- Pass 1: sum of squares reduction ---------------------
__global__ __launch_bounds__(RED_BLOCK) void clip_reduce_sq(
    const float* __restrict__ x, long long n, double* __restrict__ acc)
{
    // Per-wave private double-buffered LDS staging (no barriers needed).
    __shared__ v4f stage[RED_WAVES][2][32];

    const int  lane = threadIdx.x & 31;
    const int  wid  = threadIdx.x >> 5;
    const long long stride = (long long)gridDim.x * blockDim.x;
    const long long gid    = (long long)blockIdx.x * blockDim.x + threadIdx.x;
    const long long n4     = n >> 2;
    const v4f* __restrict__ x4 = (const v4f*)x;

    float s = 0.0f;

    // Prime the pipeline: async-copy this lane's first float4 into LDS buf 0.
    long long i   = gid;
    int       buf = 0;
    bool      have = (i < n4);
    if (have) {
        unsigned     ldsa = (unsigned)(size_t)&stage[wid][0][lane];
        const v4f*   g    = x4 + i;
        asm volatile("global_load_async_to_lds_b128 %0, %1, off"
                     :: "v"(ldsa), "v"(g) : "memory");
    }

    while (have) {
        long long inext    = i + stride;
        bool      havenext = (inext < n4);
        if (havenext) {
            // Issue the next tile into the other buffer, then wait until only
            // that newest copy can still be outstanding => current buf ready.
            unsigned   ldsa = (unsigned)(size_t)&stage[wid][buf ^ 1][lane];
            const v4f* g    = x4 + inext;
            asm volatile("global_load_async_to_lds_b128 %0, %1, off"
                         :: "v"(ldsa), "v"(g) : "memory");
            asm volatile("s_wait_asynccnt 1" ::: "memory");
        } else {
            asm volatile("s_wait_asynccnt 0" ::: "memory");
        }

        v4f v = stage[wid][buf][lane];
        s = fmaf(v.x, v.x, s);
        s = fmaf(v.y, v.y, s);
        s = fmaf(v.z, v.z, s);
        s = fmaf(v.w, v.w, s);

        buf ^= 1;
        i    = inext;
        have = havenext;
    }

    // Scalar tail (all shapes here are multiples of 4; kept for generality).
    for (long long t = (n4 << 2) + gid; t < n; t += stride)
        s = fmaf(x[t], x[t], s);

    // wave32 tree reduction.
    for (int off = 16; off > 0; off >>= 1)
        s += __shfl_down(s, off, 32);

    __shared__ float wsum[RED_WAVES];
    if (lane == 0) wsum[wid] = s;
    __syncthreads();
    if (wid == 0) {
        s = (lane < RED_WAVES) ? wsum[lane] : 0.0f;
        for (int off = RED_WAVES >> 1; off > 0; off >>= 1)
            s += __shfl_down(s, off, 32);
        if (lane == 0)
            atomicAdd(acc, (double)s);   // global_atomic_add_f64
    }
}

// ------------------------- Pass 2: scale & store ---------------------------
__global__ __launch_bounds__(RED_BLOCK) void clip_scale(
    const float* __restrict__ x, float* __restrict__ y, long long n,
    const double* __restrict__ acc)
{
    const double ssq  = *acc;
    const float  norm = sqrtf((float)ssq);
    const float  sc   = (norm > 1.0f) ? (1.0f / (norm + 1e-6f)) : 1.0f;

    const long long stride = (long long)gridDim.x * blockDim.x;
    const long long gid    = (long long)blockIdx.x * blockDim.x + threadIdx.x;
    const long long n4     = n >> 2;
    const v4f* __restrict__ x4 = (const v4f*)x;
    v4f* __restrict__       y4 = (v4f*)y;

    for (long long i = gid; i < n4; i += stride) {
        long long ip = i + 8 * stride;           // prefetch ahead (guarded)
        if (ip < n4)
            __builtin_prefetch((const void*)(x4 + ip), 0, 0);  // global_prefetch_b8
        v4f v = __builtin_nontemporal_load(x4 + i);
        v *= sc;
        __builtin_nontemporal_store(v, y4 + i);
    }
    for (long long t = (n4 << 2) + gid; t < n; t += stride)
        y[t] = x[t] * sc;
}

// ------------------------------ host side ----------------------------------
static inline int grid_for(long long work_items, long long cap)
{
    long long nb = (work_items + RED_BLOCK - 1) / RED_BLOCK;
    if (nb < 1) nb = 1;
    if (nb > cap) nb = cap;
    return (int)nb;
}

extern "C" void kernel_launch(void* const* d_in, const int* in_sizes, int n_in,
                              void* d_out, int out_size, void* d_ws, size_t ws_size,
                              hipStream_t stream)
{
    double* acc = (double*)d_ws;
    hipMemsetAsync(d_ws, 0, sizeof(double), stream);   // graph-capture safe

    // Pass 1: global sum of squares across all tensors.
    for (int t = 0; t < n_in; ++t) {
        const float* x = (const float*)d_in[t];
        long long n  = (long long)in_sizes[t];
        int grid = grid_for(n >> 2, 2048);
        clip_reduce_sq<<<grid, RED_BLOCK, 0, stream>>>(x, n, acc);
    }

    // Pass 2: scale each tensor into its slice of the concatenated output.
    float* out = (float*)d_out;
    long long off = 0;
    for (int t = 0; t < n_in; ++t) {
        const float* x = (const float*)d_in[t];
        long long n  = (long long)in_sizes[t];
        int grid = grid_for(n >> 2, 8192);
        clip_scale<<<grid, RED_BLOCK, 0, stream>>>(x, out + off, n, acc);
        off += n;
    }
    (void)out_size; (void)ws_size;
}